// GAT_24713241821269
// MI455X (gfx1250) — compile-verified
//
#include <hip/hip_runtime.h>

#define N_NODES 50000
#define N_EDGES 800000
#define TE      (N_EDGES + N_NODES)   // 850000 (with self loops)
#define IN_F    128
#define HID     256
#define OUT_F   64
#define HEADS   4
#define NEG_SLOPE 0.2f
#define BN_EPS  1e-5f

typedef __attribute__((ext_vector_type(2))) float v2f;
typedef __attribute__((ext_vector_type(8))) float v8f;

__device__ __forceinline__ v8f wmma_f32x4(v2f a, v2f b, v8f c) {
  // D = A(16x4 f32) x B(4x16 f32) + C(16x16 f32)
  return __builtin_amdgcn_wmma_f32_16x16x4_f32(false, a, false, b, (short)0, c, false, false);
}

// ---------------- GEMM1: XH[N,256] = X[N,128] @ Wgat[128,256] ----------------
// block = 128 threads (4 waves); wave w owns columns [w*64, w*64+64) of one 16-row strip.
__global__ void __launch_bounds__(128) k_gemm_xh(const float* __restrict__ X,
                                                 const float* __restrict__ W,
                                                 float* __restrict__ XH) {
  const int lane  = threadIdx.x & 31;
  const int wave  = threadIdx.x >> 5;
  const int r16   = lane & 15;
  const int kh    = lane >> 4;          // 0 or 1 (which K-pair this half-wave holds)
  const int mbase = blockIdx.x << 4;
  const int nb    = wave * 64;
  v8f acc0 = {}; v8f acc1 = {}; v8f acc2 = {}; v8f acc3 = {};
  const float* xrow = X + (size_t)(mbase + r16) * IN_F;
  for (int k0 = 0; k0 < IN_F; k0 += 4) {
    const int kb = k0 + kh * 2;
    v2f a; a.x = xrow[kb]; a.y = xrow[kb + 1];
    const float* w0 = W + (size_t)kb * HID + nb + r16;
    const float* w1 = w0 + HID;
    v2f b0; b0.x = w0[0];  b0.y = w1[0];
    v2f b1; b1.x = w0[16]; b1.y = w1[16];
    v2f b2; b2.x = w0[32]; b2.y = w1[32];
    v2f b3; b3.x = w0[48]; b3.y = w1[48];
    acc0 = wmma_f32x4(a, b0, acc0);
    acc1 = wmma_f32x4(a, b1, acc1);
    acc2 = wmma_f32x4(a, b2, acc2);
    acc3 = wmma_f32x4(a, b3, acc3);
  }
  const int rbase = mbase + kh * 8;
#pragma unroll
  for (int r = 0; r < 8; ++r) {
    float* o = XH + (size_t)(rbase + r) * HID + nb + r16;
    o[0]  = acc0[r];
    o[16] = acc1[r];
    o[32] = acc2[r];
    o[48] = acc3[r];
  }
}

// ---------------- GEMM2: H2[N,64] = Y[N,256] @ Wlin[256,64] + b ----------------
__global__ void __launch_bounds__(128) k_gemm_lin(const float* __restrict__ Y,
                                                  const float* __restrict__ W,
                                                  const float* __restrict__ bl,
                                                  float* __restrict__ H2) {
  const int lane  = threadIdx.x & 31;
  const int wave  = threadIdx.x >> 5;
  const int r16   = lane & 15;
  const int kh    = lane >> 4;
  const int mbase = blockIdx.x << 4;
  const int nb    = wave * 16;
  v8f acc = {};
  const float* yrow = Y + (size_t)(mbase + r16) * HID;
  for (int k0 = 0; k0 < HID; k0 += 4) {
    const int kb = k0 + kh * 2;
    v2f a; a.x = yrow[kb]; a.y = yrow[kb + 1];
    v2f b; b.x = W[(size_t)kb * OUT_F + nb + r16];
           b.y = W[(size_t)(kb + 1) * OUT_F + nb + r16];
    acc = wmma_f32x4(a, b, acc);
  }
  const float bias = bl[nb + r16];
  const int rbase = mbase + kh * 8;
#pragma unroll
  for (int r = 0; r < 8; ++r)
    H2[(size_t)(rbase + r) * OUT_F + nb + r16] = acc[r] + bias;
}

// ---------------- attention scalar coefficients a_s, a_d ----------------
__global__ void k_att(const float* __restrict__ XH, const float* __restrict__ ATS,
                      const float* __restrict__ ATD, float* __restrict__ AS,
                      float* __restrict__ AD) {
  int t = blockIdx.x * blockDim.x + threadIdx.x;   // t = n*4 + h
  if (t >= N_NODES * HEADS) return;
  int n = t >> 2, h = t & 3;
  const float* row = XH + (size_t)n * HID + h * OUT_F;
  const float* as = ATS + h * OUT_F;
  const float* ad = ATD + h * OUT_F;
  float s = 0.f, d = 0.f;
#pragma unroll 8
  for (int o = 0; o < OUT_F; ++o) { float v = row[o]; s += v * as[o]; d += v * ad[o]; }
  AS[t] = s; AD[t] = d;
}

__device__ __forceinline__ void edge_nodes(const int* __restrict__ ei, int e, int& s, int& d) {
  if (e < N_EDGES) { s = ei[e]; d = ei[N_EDGES + e]; }
  else             { s = e - N_EDGES; d = s; }        // self loops
}

__device__ __forceinline__ void atomicMaxF(float* addr, float v) {
  if (v >= 0.f) atomicMax((int*)addr, __float_as_int(v));
  else          atomicMin((unsigned int*)addr, (unsigned int)__float_as_int(v));
}

// ---------------- edge logits + segment max ----------------
__global__ void k_edge_logit(const int* __restrict__ ei, const float* __restrict__ AS,
                             const float* __restrict__ AD, float* __restrict__ EB,
                             float* __restrict__ EMX) {
  int t = blockIdx.x * blockDim.x + threadIdx.x;   // t = e*4 + h
  if (t >= TE * HEADS) return;
  int e = t >> 2, h = t & 3;
  int s, d; edge_nodes(ei, e, s, d);
  float v = AS[s * 4 + h] + AD[d * 4 + h];
  v = v > 0.f ? v : NEG_SLOPE * v;                  // leaky relu
  EB[t] = v;
  atomicMaxF(&EMX[d * 4 + h], v);
}

// ---------------- exp(e - max) + segment sum ----------------
__global__ void k_edge_exp(const int* __restrict__ ei, float* __restrict__ EB,
                           const float* __restrict__ EMX, float* __restrict__ DEN) {
  int t = blockIdx.x * blockDim.x + threadIdx.x;
  if (t >= TE * HEADS) return;
  int e = t >> 2, h = t & 3;
  int s, d; edge_nodes(ei, e, s, d);
  (void)s;
  float ee = __expf(EB[t] - EMX[d * 4 + h]);
  EB[t] = ee;
  atomicAdd(&DEN[d * 4 + h], ee);
}

// ---------------- weighted message scatter: one block per edge ----------------
__global__ void __launch_bounds__(256) k_aggregate(const int* __restrict__ ei,
                                                   const float* __restrict__ XH,
                                                   const float* __restrict__ EB,
                                                   const float* __restrict__ DEN,
                                                   float* __restrict__ AGG) {
  const int e = blockIdx.x;
  const int c = threadIdx.x;        // channel 0..255, head = c>>6
  const int h = c >> 6;
  int s, d; edge_nodes(ei, e, s, d);
  float alpha = EB[e * 4 + h] / (DEN[d * 4 + h] + 1e-16f);
  atomicAdd(&AGG[(size_t)d * HID + c], XH[(size_t)s * HID + c] * alpha);
}

// ---------------- BatchNorm1 stats (256 cols) ----------------
__global__ void __launch_bounds__(256) k_bn1_stats(const float* __restrict__ Hin,
                                                   const float* __restrict__ bias,
                                                   float* __restrict__ SUM,
                                                   float* __restrict__ SSQ) {
  const int c = threadIdx.x;
  const float b = bias[c];
  float s = 0.f, q = 0.f;
  for (int r = blockIdx.x; r < N_NODES; r += gridDim.x) {
    float v = Hin[(size_t)r * HID + c] + b;
    s += v; q += v * v;
  }
  atomicAdd(&SUM[c], s);
  atomicAdd(&SSQ[c], q);
}

// ---------------- BatchNorm1 apply + ReLU (in place) ----------------
__global__ void k_bn1_apply(float* __restrict__ HB, const float* __restrict__ bias,
                            const float* __restrict__ SUM, const float* __restrict__ SSQ,
                            const float* __restrict__ g, const float* __restrict__ be) {
  size_t t = (size_t)blockIdx.x * blockDim.x + threadIdx.x;  // N*256 exact
  int c = (int)(t & (HID - 1));
  float mu  = SUM[c] * (1.f / N_NODES);
  float var = SSQ[c] * (1.f / N_NODES) - mu * mu;
  float v = (HB[t] + bias[c] - mu) * rsqrtf(var + BN_EPS) * g[c] + be[c];
  HB[t] = v > 0.f ? v : 0.f;
}

// ---------------- BatchNorm2 stats (64 cols) ----------------
__global__ void __launch_bounds__(256) k_bn2_stats(const float* __restrict__ H2,
                                                   float* __restrict__ SUM,
                                                   float* __restrict__ SSQ) {
  const int c  = threadIdx.x & 63;
  const int ro = threadIdx.x >> 6;   // 0..3
  float s = 0.f, q = 0.f;
  for (int r = blockIdx.x * 4 + ro; r < N_NODES; r += gridDim.x * 4) {
    float v = H2[(size_t)r * OUT_F + c];
    s += v; q += v * v;
  }
  atomicAdd(&SUM[c], s);
  atomicAdd(&SSQ[c], q);
}

// ---------------- BatchNorm2 apply + ReLU -> d_out ----------------
__global__ void k_bn2_apply(const float* __restrict__ H2, const float* __restrict__ SUM,
                            const float* __restrict__ SSQ, const float* __restrict__ g,
                            const float* __restrict__ be, float* __restrict__ OUT) {
  size_t t = (size_t)blockIdx.x * blockDim.x + threadIdx.x;  // N*64 exact
  int c = (int)(t & 63);
  float mu  = SUM[c] * (1.f / N_NODES);
  float var = SSQ[c] * (1.f / N_NODES) - mu * mu;
  float v = (H2[t] - mu) * rsqrtf(var + BN_EPS) * g[c] + be[c];
  OUT[t] = v > 0.f ? v : 0.f;
}

__global__ void k_fill(float* __restrict__ p, int n, float v) {
  int t = blockIdx.x * blockDim.x + threadIdx.x;
  if (t < n) p[t] = v;
}

extern "C" void kernel_launch(void* const* d_in, const int* in_sizes, int n_in,
                              void* d_out, int out_size, void* d_ws, size_t ws_size,
                              hipStream_t stream) {
  (void)in_sizes; (void)n_in; (void)out_size; (void)ws_size;
  const float* x      = (const float*)d_in[0];
  // d_in[1] = edge_attr (unused, edge_dim=None)
  const float* Wg     = (const float*)d_in[2];   // [128,4,64] == [128,256]
  const float* att_s  = (const float*)d_in[3];   // [4,64]
  const float* att_d  = (const float*)d_in[4];
  const float* bias_g = (const float*)d_in[5];   // [256]
  const float* g1     = (const float*)d_in[6];
  const float* b1     = (const float*)d_in[7];
  const float* Wl     = (const float*)d_in[8];   // [256,64]
  const float* bl     = (const float*)d_in[9];
  const float* g2     = (const float*)d_in[10];
  const float* b2     = (const float*)d_in[11];
  const int*   ei     = (const int*)d_in[12];    // [2,800000]
  float* out = (float*)d_out;

  // workspace layout (floats)
  float* XH   = (float*)d_ws;          // 12,800,000
  float* AGG  = XH   + 12800000;       // 12,800,000
  float* EB   = AGG  + 12800000;       //  3,400,000
  float* AS   = EB   + 3400000;        //    200,000
  float* AD   = AS   + 200000;         //    200,000
  float* EMX  = AD   + 200000;         //    200,000
  float* DEN  = EMX  + 200000;         //    200,000
  float* H2   = DEN  + 200000;         //  3,200,000
  float* SUM1 = H2   + 3200000;        //        256
  float* SSQ1 = SUM1 + 256;            //        256
  float* SUM2 = SSQ1 + 256;            //         64
  float* SSQ2 = SUM2 + 64;             //         64

  // per-call re-initialization (graph-capture-safe)
  hipMemsetAsync(AGG, 0, (size_t)12800000 * sizeof(float), stream);
  hipMemsetAsync(DEN, 0, (size_t)200000 * sizeof(float), stream);
  hipMemsetAsync(SUM1, 0, (size_t)(256 + 256 + 64 + 64) * sizeof(float), stream);
  k_fill<<<(200000 + 255) / 256, 256, 0, stream>>>(EMX, 200000, -3.4028235e38f);

  // 1) xh = x @ W_gat   (fp32 WMMA)
  k_gemm_xh<<<N_NODES / 16, 128, 0, stream>>>(x, Wg, XH);
  // 2) attention coefficients
  k_att<<<(N_NODES * HEADS + 255) / 256, 256, 0, stream>>>(XH, att_s, att_d, AS, AD);
  // 3) edge logits + segment max
  k_edge_logit<<<(TE * HEADS + 255) / 256, 256, 0, stream>>>(ei, AS, AD, EB, EMX);
  // 4) exp + segment sum
  k_edge_exp<<<(TE * HEADS + 255) / 256, 256, 0, stream>>>(ei, EB, EMX, DEN);
  // 5) weighted scatter-aggregate
  k_aggregate<<<TE, 256, 0, stream>>>(ei, XH, EB, DEN, AGG);
  // 6) BN1 + ReLU
  k_bn1_stats<<<200, 256, 0, stream>>>(AGG, bias_g, SUM1, SSQ1);
  k_bn1_apply<<<N_NODES, 256, 0, stream>>>(AGG, bias_g, SUM1, SSQ1, g1, b1);
  // 7) h2 = y1 @ W_lin + b   (fp32 WMMA)
  k_gemm_lin<<<N_NODES / 16, 128, 0, stream>>>(AGG, Wl, bl, H2);
  // 8) BN2 + ReLU -> out
  k_bn2_stats<<<128, 256, 0, stream>>>(H2, SUM2, SSQ2);
  k_bn2_apply<<<(N_NODES * OUT_F) / 256, 256, 0, stream>>>(H2, SUM2, SSQ2, g2, b2, out);
}